// GCNModelStrAttScatTwoDecodersVAE_481036337855
// MI455X (gfx1250) — compile-verified
//
#include <hip/hip_runtime.h>
#include <hip/hip_bf16.h>

// ---------------------------------------------------------------------------
// GAT-VAE forward, MI455X (gfx1250). fp32 WMMA 16x16x4 everywhere (memory
// bound -> exact precision is free). Adjacency is 0.01*mask; we stream a u8
// mask (64MB) instead of 268MB fp32, synthesize A-operands in-register, and
// double-buffer the K loop with zero predication (feature dim padded
// 1433->1536 so every tile is full).
// ---------------------------------------------------------------------------

#define NN    8192
#define H2D   128
#define H1D   256
#define IND   1433
#define INDP  1536      // padded feature dim: 12 groups of 128, no tails
#define EPSV  1e-5f
#define NEGV  (-9e15f)
#define SLP   0.2f
#define ADJW  0.01f

typedef __attribute__((ext_vector_type(2))) float v2f;
typedef __attribute__((ext_vector_type(8))) float v8f;

// ---------------------------------------------------------------------------
// WMMA GEMM: OUT[8192 x C] = A' @ B,  A'[8192 x K]:
//   MODE 0: A' = A (dense f32)
//   MODE 1: A' = softmax-weights synthesized from mask/f1/f2/rowstats
//   MODE 2: A' = mask (0/1), result scaled by ADJW in epilogue
// Wave tile: 16 rows x 128 cols (8 col tiles), K stepped by 4, double
// buffered. COLW=true: the 8 waves of a block take 8 consecutive column
// groups of the SAME 16 rows (mask rows shared through WGP$/L2).
// C and K must be multiples of 128/8 respectively (guaranteed by padding).
// ---------------------------------------------------------------------------
template <int MODE, bool RELU, bool COLW>
__global__ __launch_bounds__(256) void gemm16_kernel(
    const float* __restrict__ A, const unsigned char* __restrict__ Msk,
    const float* __restrict__ B, float* __restrict__ OUT, int K, int C,
    const float* __restrict__ f1v, const float* __restrict__ f2v,
    const float* __restrict__ mrow, const float* __restrict__ srow) {
  const int lane = threadIdx.x & 31;
  const int wave = threadIdx.x >> 5;
  int row0, c0;
  if (COLW) {
    row0 = blockIdx.x * 16;
    c0   = (blockIdx.y * 8 + wave) * 128;
    if (c0 >= C) return;  // no barriers in this kernel -> early exit is safe
  } else {
    row0 = (blockIdx.x * 8 + wave) * 16;
    c0   = blockIdx.y * 128;
  }
  const int l15  = lane & 15;
  const int hi   = lane >> 4;  // 0 -> K=k+0..1, 1 -> K=k+2..3
  const int arow = row0 + l15;
  const int koff = hi * 2;

  float f1r = 0.f, mr = 0.f, rs = 0.f;
  if (MODE == 1) {
    f1r = f1v[arow];
    mr  = mrow[arow];
    rs  = 1.0f / srow[arow];
  }

  struct ABuf { v2f a; v2f f2; };

  const float* __restrict__ Abase =
      (MODE == 0) ? (A + (size_t)arow * (size_t)K + koff) : nullptr;
  const unsigned char* __restrict__ Mbase =
      (MODE != 0) ? (Msk + (size_t)arow * (size_t)K + koff) : nullptr;
  const float* __restrict__ Bbase = B + c0 + l15;

  auto loadA = [&](int k) {
    ABuf r;
    if (MODE == 0) {
      r.a.x = Abase[k];
      r.a.y = Abase[k + 1];
      r.f2.x = 0.f; r.f2.y = 0.f;
    } else {
      const unsigned int mm = *(const unsigned short*)(Mbase + k);
      r.a.x = __uint_as_float(mm);
      r.a.y = 0.f;
      if (MODE == 1) {
        r.f2.x = f2v[k + koff];
        r.f2.y = f2v[k + koff + 1];
      } else {
        r.f2.x = 0.f; r.f2.y = 0.f;
      }
    }
    return r;
  };

  auto loadB = [&](int k, v2f* b) {
    const float* __restrict__ p = Bbase + (size_t)(k + koff) * (size_t)C;
#pragma unroll
    for (int t = 0; t < 8; ++t) {
      b[t].x = p[t * 16];       // row k+koff
      b[t].y = p[t * 16 + C];   // row k+koff+1
    }
  };

  v8f acc[8];
#pragma unroll
  for (int t = 0; t < 8; ++t)
#pragma unroll
    for (int r = 0; r < 8; ++r) acc[t][r] = 0.0f;

  auto compute = [&](const ABuf& ab, const v2f* b) {
    v2f a;
    if (MODE == 0) {
      a = ab.a;
    } else if (MODE == 2) {
      const unsigned int mm = __float_as_uint(ab.a.x);
      a.x = (mm & 0x00ffu) ? 1.0f : 0.0f;
      a.y = (mm & 0xff00u) ? 1.0f : 0.0f;
    } else {  // MODE 1: softmax(where(mask, leaky_relu(f1_i + f2_j), NEG))
      const unsigned int mm = __float_as_uint(ab.a.x);
      float e0 = f1r + ab.f2.x;
      float e1 = f1r + ab.f2.y;
      e0 = e0 > 0.f ? e0 : SLP * e0;
      e1 = e1 > 0.f ? e1 : SLP * e1;
      const float v0 = (mm & 0x00ffu) ? e0 : NEGV;
      const float v1 = (mm & 0xff00u) ? e1 : NEGV;
      a.x = __expf(v0 - mr) * rs;
      a.y = __expf(v1 - mr) * rs;
    }
#pragma unroll
    for (int t = 0; t < 8; ++t)
      acc[t] = __builtin_amdgcn_wmma_f32_16x16x4_f32(
          false, a, false, b[t], (short)0, acc[t], false, false);
  };

  // software-pipelined K loop (K % 8 == 0), ping-pong buffers
  ABuf a0, a1;
  v2f b0[8], b1[8];
  a0 = loadA(0);
  loadB(0, b0);
  int k = 0;
  for (; k + 8 < K; k += 8) {
    a1 = loadA(k + 4);
    loadB(k + 4, b1);
    compute(a0, b0);
    a0 = loadA(k + 8);
    loadB(k + 8, b0);
    compute(a1, b1);
  }
  a1 = loadA(K - 4);
  loadB(K - 4, b1);
  compute(a0, b0);
  compute(a1, b1);

  // epilogue: VGPR r, lanes 0-15 -> M=r, N=lane; lanes 16-31 -> M=8+r
#pragma unroll
  for (int t = 0; t < 8; ++t) {
    const int c = c0 + t * 16 + l15;
#pragma unroll
    for (int r = 0; r < 8; ++r) {
      float v = acc[t][r];
      if (MODE == 2) v *= ADJW;           // adj = ADJW * mask
      if (RELU) v = v > 0.f ? v : 0.f;
      OUT[(size_t)(row0 + r + hi * 8) * (size_t)C + c] = v;
    }
  }
}

// ---------------------------------------------------------------------------
// u8 mask = (adj > 0), 4 elements per thread
// ---------------------------------------------------------------------------
__global__ __launch_bounds__(256) void mask_kernel(
    const float* __restrict__ adj, unsigned char* __restrict__ msk) {
  const size_t i = (size_t)blockIdx.x * blockDim.x + threadIdx.x;
  const float4 v = ((const float4*)adj)[i];
  uchar4 m;
  m.x = v.x > 0.f ? 1 : 0;
  m.y = v.y > 0.f ? 1 : 0;
  m.z = v.z > 0.f ? 1 : 0;
  m.w = v.w > 0.f ? 1 : 0;
  ((uchar4*)msk)[i] = m;
}

// ---------------------------------------------------------------------------
// W_fd2 [256 x 1433] -> zero-padded [256 x 1536]
// ---------------------------------------------------------------------------
__global__ __launch_bounds__(256) void padw_kernel(
    const float* __restrict__ W, float* __restrict__ Wp) {
  const int idx = blockIdx.x * 256 + threadIdx.x;  // over 256*1536
  const int r = idx / INDP, c = idx - r * INDP;
  Wp[idx] = (c < IND) ? W[r * IND + c] : 0.f;
}

// ---------------------------------------------------------------------------
// f1 = Wh @ a[:128], f2 = Wh @ a[128:]  (one wave per row)
// ---------------------------------------------------------------------------
__global__ __launch_bounds__(256) void f12_kernel(
    const float* __restrict__ Wh, const float* __restrict__ a_att,
    float* __restrict__ f1, float* __restrict__ f2) {
  const int lane = threadIdx.x & 31;
  const int wave = threadIdx.x >> 5;
  const int row  = blockIdx.x * 8 + wave;
  float s1 = 0.f, s2 = 0.f;
#pragma unroll
  for (int t = 0; t < 4; ++t) {
    const int c = lane + 32 * t;
    const float w = Wh[(size_t)row * H2D + c];
    s1 += w * a_att[c];
    s2 += w * a_att[H2D + c];
  }
#pragma unroll
  for (int off = 16; off > 0; off >>= 1) {
    s1 += __shfl_down(s1, off, 32);
    s2 += __shfl_down(s2, off, 32);
  }
  if (lane == 0) {
    f1[row] = s1;
    f2[row] = s2;
  }
}

// ---------------------------------------------------------------------------
// Per-row online softmax stats over masked leaky-relu logits (flash style).
// Streams the u8 mask with uchar4 loads, one wave per row.
// ---------------------------------------------------------------------------
__global__ __launch_bounds__(256) void att_stats_kernel(
    const unsigned char* __restrict__ msk, const float* __restrict__ f1,
    const float* __restrict__ f2, float* __restrict__ mout,
    float* __restrict__ sout) {
  const int lane = threadIdx.x & 31;
  const int wave = threadIdx.x >> 5;
  const int row  = blockIdx.x * 8 + wave;
  const float f1r = f1[row];
  float m = -3.0e38f, s = 0.f;
  auto upd = [&](unsigned char mk, float f2j) {
    float e = f1r + f2j;
    e = e > 0.f ? e : SLP * e;
    const float v = mk ? e : NEGV;
    if (v > m) {
      s = s * __expf(m - v) + 1.0f;
      m = v;
    } else {
      s += __expf(v - m);
    }
  };
  for (int jb = lane * 4; jb < NN; jb += 128) {
    const uchar4 mk = *(const uchar4*)(msk + (size_t)row * NN + jb);
    const float4 ff = *(const float4*)(f2 + jb);
    upd(mk.x, ff.x);
    upd(mk.y, ff.y);
    upd(mk.z, ff.z);
    upd(mk.w, ff.w);
  }
#pragma unroll
  for (int off = 16; off > 0; off >>= 1) {
    const float m2 = __shfl_xor(m, off, 32);
    const float s2 = __shfl_xor(s, off, 32);
    const float M  = fmaxf(m, m2);
    s = s * __expf(m - M) + s2 * __expf(m2 - M);
    m = M;
  }
  if (lane == 0) {
    mout[row] = m;
    sout[row] = s;
  }
}

// ---------------------------------------------------------------------------
// BatchNorm1d training forward (biased var); strided so padded buffers can
// feed the unpadded output. Safe for dst == src when strides match.
// ---------------------------------------------------------------------------
__global__ __launch_bounds__(256) void bn_kernel(
    const float* __restrict__ src, const float* __restrict__ g,
    const float* __restrict__ b, float* __restrict__ dst, int C, int ss,
    int ds) {
  const int col = blockIdx.x * blockDim.x + threadIdx.x;
  if (col >= C) return;
  float sum = 0.f, sq = 0.f;
  for (int r = 0; r < NN; ++r) {
    const float v = src[(size_t)r * (size_t)ss + col];
    sum += v;
    sq += v * v;
  }
  const float inv_n = 1.0f / (float)NN;
  const float mean  = sum * inv_n;
  float var = sq * inv_n - mean * mean;
  var = var > 0.f ? var : 0.f;
  const float scale = rsqrtf(var + EPSV) * g[col];
  const float shift = b[col] - mean * scale;
  for (int r = 0; r < NN; ++r)
    dst[(size_t)r * (size_t)ds + col] =
        src[(size_t)r * (size_t)ss + col] * scale + shift;
}

// ---------------------------------------------------------------------------
// Tiled transpose s1[8192 x 256] -> s1T[256 x 8192]
// ---------------------------------------------------------------------------
__global__ void transpose_kernel(const float* __restrict__ src,
                                 float* __restrict__ dst) {
  __shared__ float tile[32][33];
  const int x = blockIdx.x * 32 + threadIdx.x;
  const int y = blockIdx.y * 32 + threadIdx.y;
#pragma unroll
  for (int j = 0; j < 32; j += 8)
    tile[threadIdx.y + j][threadIdx.x] = src[(size_t)(y + j) * H1D + x];
  __syncthreads();
  const int ox = blockIdx.y * 32 + threadIdx.x;
  const int oy = blockIdx.x * 32 + threadIdx.y;
#pragma unroll
  for (int j = 0; j < 32; j += 8)
    dst[(size_t)(oy + j) * NN + ox] = tile[threadIdx.x][threadIdx.y + j];
}

// ---------------------------------------------------------------------------
extern "C" void kernel_launch(void* const* d_in, const int* in_sizes, int n_in,
                              void* d_out, int out_size, void* d_ws,
                              size_t ws_size, hipStream_t stream) {
  (void)in_sizes; (void)n_in; (void)out_size; (void)ws_size;

  const float* x     = (const float*)d_in[0];
  const float* adj   = (const float*)d_in[1];
  const float* W_att = (const float*)d_in[2];
  const float* a_att = (const float*)d_in[3];
  const float* W_fd1 = (const float*)d_in[4];
  const float* bn1_g = (const float*)d_in[5];
  const float* bn1_b = (const float*)d_in[6];
  const float* W_fd2 = (const float*)d_in[7];
  const float* bn2_g = (const float*)d_in[8];
  const float* bn2_b = (const float*)d_in[9];
  const float* W_sd1 = (const float*)d_in[10];
  const float* bn3_g = (const float*)d_in[11];
  const float* bn3_b = (const float*)d_in[12];

  // ---- workspace layout ----
  char* wsb = (char*)d_ws;
  constexpr size_t SZ_WH   = (size_t)NN * H2D * sizeof(float);    // 4 MiB
  constexpr size_t SZ_VEC  = (size_t)NN * sizeof(float);          // 32 KiB
  constexpr size_t SZ_WP   = (size_t)H1D * INDP * sizeof(float);  // 1.5 MiB
  constexpr size_t SZ_MSK  = (size_t)NN * NN;                     // 64 MiB
  constexpr size_t SZ_BIG  = (size_t)NN * INDP * sizeof(float);   // 48 MiB
  float* Wh    = (float*)(wsb);
  float* z     = (float*)(wsb + SZ_WH);
  float* f1    = (float*)(wsb + 2 * SZ_WH);
  float* f2    = (float*)(wsb + 2 * SZ_WH + SZ_VEC);
  float* mrow  = (float*)(wsb + 2 * SZ_WH + 2 * SZ_VEC);
  float* srow  = (float*)(wsb + 2 * SZ_WH + 3 * SZ_VEC);
  float* Wfd2p = (float*)(wsb + 2 * SZ_WH + 4 * SZ_VEC);
  unsigned char* msk = (unsigned char*)(wsb + 2 * SZ_WH + 4 * SZ_VEC + SZ_WP);
  float* bufB = (float*)(wsb + 2 * SZ_WH + 4 * SZ_VEC + SZ_WP + SZ_MSK);
  float* bufC = (float*)(wsb + 2 * SZ_WH + 4 * SZ_VEC + SZ_WP + SZ_MSK + SZ_BIG);

  float* feat    = (float*)d_out;                     // [8192 x 1433]
  float* structo = (float*)d_out + (size_t)NN * IND;  // [8192 x 8192]

  const dim3 blk(256);
  const dim3 gRows(NN / 8);  // f12 / att_stats: 8 waves/block, 1 row/wave
  auto gridRowW = [](int C) { return dim3(NN / 128, C / 128); };      // COLW=0
  auto gridColW = [](int C) { return dim3(NN / 16, (C + 1023) / 1024); }; // COLW=1

  // 0) adjacency mask (u8) + padded W_fd2
  mask_kernel<<<dim3((NN * (size_t)NN / 4) / 256), blk, 0, stream>>>(adj, msk);
  padw_kernel<<<dim3(H1D * INDP / 256), blk, 0, stream>>>(W_fd2, Wfd2p);

  // 1) Wh = x @ W_att                         [8192 x 128]
  gemm16_kernel<0, false, false><<<gridRowW(H2D), blk, 0, stream>>>(
      x, nullptr, W_att, Wh, H2D, H2D, nullptr, nullptr, nullptr, nullptr);
  // 2) f1/f2 row dots
  f12_kernel<<<gRows, blk, 0, stream>>>(Wh, a_att, f1, f2);
  // 3) per-row softmax stats
  att_stats_kernel<<<gRows, blk, 0, stream>>>(msk, f1, f2, mrow, srow);
  // 4) z = att @ Wh (attention weights synthesized from mask)
  gemm16_kernel<1, false, true><<<gridColW(H2D), blk, 0, stream>>>(
      nullptr, msk, Wh, z, NN, H2D, f1, f2, mrow, srow);

  // ---- feature decoder ----
  // 5) t1 = z @ W_fd1                         [8192 x 256]
  gemm16_kernel<0, false, false><<<gridRowW(H1D), blk, 0, stream>>>(
      z, nullptr, W_fd1, bufB, H2D, H1D, nullptr, nullptr, nullptr, nullptr);
  // 6) h1 = relu(adj @ t1) = ADJW*relu(mask @ t1)
  gemm16_kernel<2, true, true><<<gridColW(H1D), blk, 0, stream>>>(
      nullptr, msk, bufB, bufC, NN, H1D, nullptr, nullptr, nullptr, nullptr);
  // 7) bn1 (in place)
  bn_kernel<<<dim3((H1D + 255) / 256), blk, 0, stream>>>(
      bufC, bn1_g, bn1_b, bufC, H1D, H1D, H1D);
  // 8) t2 = h1 @ W_fd2pad                     [8192 x 1536]
  gemm16_kernel<0, false, false><<<gridRowW(INDP), blk, 0, stream>>>(
      bufC, nullptr, Wfd2p, bufB, H1D, INDP, nullptr, nullptr, nullptr,
      nullptr);
  // 9) featpre = relu(adj @ t2)               [8192 x 1536]
  gemm16_kernel<2, true, true><<<gridColW(INDP), blk, 0, stream>>>(
      nullptr, msk, bufB, bufC, NN, INDP, nullptr, nullptr, nullptr, nullptr);
  // 10) bn2 -> feat output (strided 1536 -> 1433)
  bn_kernel<<<dim3((IND + 255) / 256), blk, 0, stream>>>(
      bufC, bn2_g, bn2_b, feat, IND, INDP, IND);

  // ---- structure decoder ----
  // 11) t3 = z @ W_sd1                        [8192 x 256]
  gemm16_kernel<0, false, false><<<gridRowW(H1D), blk, 0, stream>>>(
      z, nullptr, W_sd1, bufB, H2D, H1D, nullptr, nullptr, nullptr, nullptr);
  // 12) s1 = relu(adj @ t3)
  gemm16_kernel<2, true, true><<<gridColW(H1D), blk, 0, stream>>>(
      nullptr, msk, bufB, bufC, NN, H1D, nullptr, nullptr, nullptr, nullptr);
  // 13) bn3 (in place)
  bn_kernel<<<dim3((H1D + 255) / 256), blk, 0, stream>>>(
      bufC, bn3_g, bn3_b, bufC, H1D, H1D, H1D);
  // 14) s1T = transpose(s1)                   [256 x 8192]
  transpose_kernel<<<dim3(H1D / 32, NN / 32), dim3(32, 8), 0, stream>>>(
      bufC, bufB);
  // 15) struct = s1 @ s1T                     [8192 x 8192]
  gemm16_kernel<0, false, false><<<gridRowW(NN), blk, 0, stream>>>(
      bufC, nullptr, bufB, structo, H1D, NN, nullptr, nullptr, nullptr,
      nullptr);
}